// Topk_search_rank_11381663334689
// MI455X (gfx1250) — compile-verified
//
#include <hip/hip_runtime.h>
#include <math.h>

#define B_  32
#define N_  8192
#define D_  128
#define K_  64

typedef __attribute__((ext_vector_type(2))) float v2f;
typedef __attribute__((ext_vector_type(8))) float v8f;

// ---------------- Kernel 1: normalize modal1 (eps 1e-12) ----------------
__global__ __launch_bounds__(128) void k_norm_m1(const float* __restrict__ m1,
                                                 float* __restrict__ m1n) {
  __shared__ float red[128];
  const int b = blockIdx.x, t = threadIdx.x;
  const float v = m1[b * D_ + t];
  red[t] = v * v;
  __syncthreads();
  for (int s = 64; s > 0; s >>= 1) {
    if (t < s) red[t] += red[t + s];
    __syncthreads();
  }
  const float inv = 1.0f / fmaxf(sqrtf(red[0]), 1e-12f);
  m1n[b * D_ + t] = v * inv;
}

// ---------------- Kernel 2: streaming dot + row-norm via fp32 WMMA ------
// One wave handles a 16-row tile of modal2 for one batch.
// A = m2 tile rows (16 x 4 fp32 chunk), B = m1n chunk broadcast to all 16
// columns, C accumulates over 32 chunks (K = 128).
__global__ __launch_bounds__(256) void k_cos(const float* __restrict__ m2,
                                             const float* __restrict__ m1n,
                                             float* __restrict__ dots,
                                             float* __restrict__ invn) {
  const int lane = threadIdx.x & 31;
  const int wave = threadIdx.x >> 5;
  const int tile = blockIdx.x * 8 + wave;     // B_*N_/16 = 16384 tiles
  const int b    = tile >> 9;                 // 512 tiles per batch
  const int n0   = (tile & 511) << 4;
  const int row  = lane & 15;
  const int half = lane >> 4;                 // 0: K={0,1}, 1: K={2,3}

  const float* __restrict__ rp = m2 + ((size_t)b * N_ + (size_t)(n0 + row)) * D_ + 2 * half;
  const float* __restrict__ mp = m1n + b * D_ + 2 * half;

  v8f c = {0.f, 0.f, 0.f, 0.f, 0.f, 0.f, 0.f, 0.f};
  float nsq = 0.f;
#pragma unroll
  for (int d = 0; d < D_; d += 4) {
    const float2 a2 = *reinterpret_cast<const float2*>(rp + d);
    const float2 b2 = *reinterpret_cast<const float2*>(mp + d);
    nsq = fmaf(a2.x, a2.x, nsq);
    nsq = fmaf(a2.y, a2.y, nsq);
    v2f av; av.x = a2.x; av.y = a2.y;
    v2f bv; bv.x = b2.x; bv.y = b2.y;
    c = __builtin_amdgcn_wmma_f32_16x16x4_f32(false, av, false, bv,
                                              (short)0, c, false, false);
  }

  // combine the two half-lane partial sums of squares for this row (wave32)
  nsq += __shfl_xor(nsq, 16, 32);
  if (lane < 16) {
    invn[(size_t)b * N_ + n0 + lane] = 1.0f / fmaxf(sqrtf(nsq), 1e-12f);
  }

  // C/D layout: lanes 0-15 hold rows 0..7 in c[0..7]; lanes 16-31 rows 8..15.
  // All 16 columns are identical, so column 0 of each half suffices.
  if ((lane & 15) == 0) {
    float* dp = dots + (size_t)b * N_ + n0 + half * 8;
#pragma unroll
    for (int r = 0; r < 8; ++r) dp[r] = c[r];
  }
}

// ---------------- Kernel 3: per-batch top-K (K=64) ----------------------
__global__ __launch_bounds__(256) void k_topk(const float* __restrict__ dots,
                                              const float* __restrict__ invn,
                                              int* __restrict__ idx) {
  __shared__ float vals[N_];     // 32 KB
  __shared__ float bv[256];
  __shared__ int   bi[256];
  const int b = blockIdx.x, t = threadIdx.x;
  for (int i = t; i < N_; i += 256)
    vals[i] = dots[(size_t)b * N_ + i] * invn[(size_t)b * N_ + i];
  __syncthreads();

  for (int k = 0; k < K_; ++k) {
    float best = -3.402823466e38f;
    int besti = N_;
    for (int i = t; i < N_; i += 256) {
      const float v = vals[i];
      if (v > best || (v == best && i < besti)) { best = v; besti = i; }
    }
    bv[t] = best; bi[t] = besti;
    __syncthreads();
    for (int s = 128; s > 0; s >>= 1) {
      if (t < s) {
        if (bv[t + s] > bv[t] || (bv[t + s] == bv[t] && bi[t + s] < bi[t])) {
          bv[t] = bv[t + s]; bi[t] = bi[t + s];
        }
      }
      __syncthreads();
    }
    if (t == 0) { idx[b * K_ + k] = bi[0]; vals[bi[0]] = -3.402823466e38f; }
    __syncthreads();
  }
}

// ---------------- Kernel 4: softmax + gather + weighted sum -------------
__global__ __launch_bounds__(128) void k_combine(const float* __restrict__ m2,
                                                 const float* __restrict__ w,
                                                 const int* __restrict__ idx,
                                                 float* __restrict__ out) {
  __shared__ float sw[K_];
  __shared__ float red[128];
  __shared__ float sinv;
  const int b = blockIdx.x, t = threadIdx.x;

  if (t < K_) sw[t] = w[b * K_ + t];
  __syncthreads();
  if (t == 0) {
    float m = -3.402823466e38f;
    for (int k = 0; k < K_; ++k) m = fmaxf(m, sw[k]);
    float s = 0.f;
    for (int k = 0; k < K_; ++k) { sw[k] = expf(sw[k] - m); s += sw[k]; }
    const float is = 1.f / s;
    for (int k = 0; k < K_; ++k) sw[k] *= is;
  }
  __syncthreads();

  float acc = 0.f;
  for (int k = 0; k < K_; ++k) {
    const int n = idx[b * K_ + k];
    const float v = m2[((size_t)b * N_ + (size_t)n) * D_ + t];
    red[t] = v * v;
    __syncthreads();
    for (int s = 64; s > 0; s >>= 1) {
      if (t < s) red[t] += red[t + s];
      __syncthreads();
    }
    if (t == 0) sinv = 1.f / fmaxf(sqrtf(red[0]), 1e-12f);
    __syncthreads();
    acc += sw[k] * v * sinv;
    __syncthreads();
  }
  out[b * D_ + t] = acc;
}

// ------------------------------------------------------------------------
extern "C" void kernel_launch(void* const* d_in, const int* in_sizes, int n_in,
                              void* d_out, int out_size, void* d_ws, size_t ws_size,
                              hipStream_t stream) {
  (void)in_sizes; (void)n_in; (void)out_size; (void)ws_size;
  const float* m1 = (const float*)d_in[0];   // [B, D]
  const float* m2 = (const float*)d_in[1];   // [B, N, D]
  const float* w  = (const float*)d_in[2];   // [B, K]
  float* out = (float*)d_out;                // [B, D]

  float* ws   = (float*)d_ws;
  float* m1n  = ws;                          // B*D
  float* dots = m1n + B_ * D_;               // B*N
  float* invn = dots + (size_t)B_ * N_;      // B*N
  int*   idx  = (int*)(invn + (size_t)B_ * N_); // B*K

  k_norm_m1<<<B_, D_, 0, stream>>>(m1, m1n);
  k_cos<<<(B_ * N_ / 16) / 8, 256, 0, stream>>>(m2, m1n, dots, invn);
  k_topk<<<B_, 256, 0, stream>>>(dots, invn, idx);
  k_combine<<<B_, D_, 0, stream>>>(m2, w, idx, out);
}